// MultiHeadedCrossSliceAttentionModule_16183436771942
// MI455X (gfx1250) — compile-verified
//
#include <hip/hip_runtime.h>
#include <math.h>

// ---------------------------------------------------------------------------
// Sizes (compile-time, from the reference)
// ---------------------------------------------------------------------------
#define BB      20
#define CC      64
#define HH      128
#define WW      128
#define HEADS   3
#define PHW     1024        // 32*32 pooled spatial
#define HW      16384       // 128*128 spatial
#define DDIM    65536       // C * PHW
#define NTOT    20971520    // B*C*H*W
#define NPOOL   1310720     // B*C*32*32
#define EPSLN   1e-5f

typedef float v2f_ __attribute__((ext_vector_type(2)));
typedef float v8f_ __attribute__((ext_vector_type(8)));

// ---------------------------------------------------------------------------
// Wave-level 16x16 output tile, K=64, fp32 WMMA (V_WMMA_F32_16X16X4_F32).
// A[m][k] at A[m*sa+k] (16x64 slice), B[k][n] at B[k*sb+n] (64x16 slice).
// Lane layout per CDNA5 ISA 7.12.2:
//   A 16x4: lanes 0-15 hold K=0,1 ; lanes 16-31 hold K=2,3 (M = lane%16)
//   B 4x16: lanes 0-15 hold K=0,1 rows ; lanes 16-31 hold K=2,3 (N = lane%16)
//   C/D   : VGPR r -> M = r + 8*(lane/16), N = lane%16
// ---------------------------------------------------------------------------
__device__ __forceinline__ v8f_ wmma64(const float* __restrict__ A, int sa,
                                       const float* __restrict__ B, int sb,
                                       v8f_ acc, int lm, int half) {
#pragma unroll
  for (int k0 = 0; k0 < 64; k0 += 4) {
    const int koff = k0 + 2 * half;
    v2f_ a; a[0] = A[lm * sa + koff];     a[1] = A[lm * sa + koff + 1];
    v2f_ b; b[0] = B[koff * sb + lm];     b[1] = B[(koff + 1) * sb + lm];
    acc = __builtin_amdgcn_wmma_f32_16x16x4_f32(false, a, false, b,
                                                (short)0, acc, false, false);
  }
  return acc;
}

__device__ __forceinline__ float gelu_exact(float x) {
  return 0.5f * x * (1.0f + erff(x * 0.70710678118654752f));
}

// ---------------------------------------------------------------------------
// Kernel 1: q,k = per-head 64x64 channel matmul on pooled features (WMMA)
// q[h,b,o,xy] = sum_c Wq[h,o,c] * pooled[b,c,xy]
// grid = 2(sel) * 3(h) * 20(b) * 32 sub-blocks ; block = 256 (8 waves)
// each wave: one 16x16 tile; 256 tiles per (sel,h,b)
// ---------------------------------------------------------------------------
__global__ __launch_bounds__(256) void qk_kernel(const float* __restrict__ pooled,
                                                 const float* __restrict__ Wq,
                                                 const float* __restrict__ Wk,
                                                 float* __restrict__ qbuf,
                                                 float* __restrict__ kbuf) {
  const int tid  = threadIdx.x;
  const int lane = tid & 31, wv = tid >> 5;
  const int lm   = lane & 15, half = lane >> 4;
  const int blk  = blockIdx.x;
  const int sub  = blk & 31;
  const int grp  = blk >> 5;           // 0..119
  const int sel  = grp / 60;           // 0 = q, 1 = k
  const int hb   = grp % 60;
  const int h = hb / 20, b = hb % 20;
  const int tile = sub * 8 + wv;       // 0..255
  const int ot   = tile >> 6;          // 0..3  (16 out-channels each)
  const int xyt  = tile & 63;          // 0..63 (16 cols each)

  const float* Wm = (sel ? Wk : Wq) + h * 4096 + (ot * 16) * 64;
  const float* Bp = pooled + b * (CC * PHW) + xyt * 16;

  v8f_ acc = {};
  acc = wmma64(Wm, 64, Bp, PHW, acc, lm, half);

  float* dst = (sel ? kbuf : qbuf) +
               ((size_t)(h * 20 + b) * 64 + ot * 16) * PHW + xyt * 16;
#pragma unroll
  for (int r = 0; r < 8; ++r)
    dst[(r + 8 * half) * PHW + lm] = acc[r];
}

// ---------------------------------------------------------------------------
// Kernel 2: scores[h,b,e] = dot(q[h,b,:], k[h,e,:]) / 256   (d = 65536)
// ---------------------------------------------------------------------------
__global__ __launch_bounds__(256) void scores_kernel(const float* __restrict__ qb,
                                                     const float* __restrict__ kb,
                                                     float* __restrict__ scores) {
  __shared__ float red[256];
  const int tid = threadIdx.x;
  const int blk = blockIdx.x;          // h*400 + b*20 + e
  const int h = blk / 400, rem = blk % 400, b = rem / 20, e = rem % 20;
  const float* qp = qb + (size_t)(h * 20 + b) * DDIM;
  const float* kp = kb + (size_t)(h * 20 + e) * DDIM;
  float s = 0.f;
  for (int i = tid; i < DDIM; i += 256) s += qp[i] * kp[i];
  red[tid] = s; __syncthreads();
  for (int off = 128; off; off >>= 1) {
    if (tid < off) red[tid] += red[tid + off];
    __syncthreads();
  }
  if (tid == 0) scores[blk] = red[0] * (1.0f / 256.0f);
}

// ---------------------------------------------------------------------------
// Kernel 3: softmax over e (last axis). 60 rows of 20. One wave per row.
// ---------------------------------------------------------------------------
__global__ __launch_bounds__(32) void softmax_kernel(const float* __restrict__ scores,
                                                     float* __restrict__ attn) {
  const int blk = blockIdx.x;          // h*20 + b
  const int lane = threadIdx.x;
  float s = (lane < 20) ? scores[blk * 20 + lane] : -3.0e38f;
  float m = s;
  for (int mask = 16; mask; mask >>= 1) m = fmaxf(m, __shfl_xor(m, mask));
  float e = (lane < 20) ? expf(s - m) : 0.f;
  float sum = e;
  for (int mask = 16; mask; mask >>= 1) sum += __shfl_xor(sum, mask);
  if (lane < 20) attn[blk * 20 + lane] = e / sum;
}

// ---------------------------------------------------------------------------
// Kernel 4: WA[h][o][c'] = sum_c W1[o, h*64+c] * Wv[h,c,c']   (3 x 64x64)
// ---------------------------------------------------------------------------
__global__ __launch_bounds__(256) void wa_kernel(const float* __restrict__ W1,
                                                 const float* __restrict__ Wv,
                                                 float* __restrict__ WA) {
  const int t = blockIdx.x * 256 + threadIdx.x;
  if (t < HEADS * 64 * 64) {
    const int h = t / 4096, o = (t >> 6) & 63, cp = t & 63;
    float s = 0.f;
    for (int c = 0; c < 64; ++c)
      s += W1[o * 192 + h * 64 + c] * Wv[(h * 64 + c) * 64 + cp];
    WA[t] = s;
  }
}

// ---------------------------------------------------------------------------
// Kernel 5 (main fusion): per (b, 64-col xy tile):
//   fmix[h,c',j] = sum_e attn[h,b,e] * features[e,c',xy]     (batch mix, L2-hot)
//   out1[o,j]    = sum_h WA[h] @ fmix[h]  + b1               (WMMA, K=192)
//   x1           = gelu(out1) + features[b]                  -> X (scratch)
//   block-level partial sums (for global LayerNorm-1)
// ---------------------------------------------------------------------------
__global__ __launch_bounds__(256) void fused1_kernel(const float* __restrict__ features,
                                                     const float* __restrict__ attn,
                                                     const float* __restrict__ WA,
                                                     const float* __restrict__ b1,
                                                     float* __restrict__ X,
                                                     float* __restrict__ partials) {
  __shared__ float fmix[3][64][64];    // 48 KB
  __shared__ float attn_s[3][20];
  __shared__ float rs[256], rq[256];
  const int tid = threadIdx.x;
  const int b   = blockIdx.x / 256;
  const int xy0 = (blockIdx.x % 256) * 64;

  if (tid < 60)
    attn_s[tid / 20][tid % 20] = attn[(tid / 20) * 400 + b * 20 + (tid % 20)];
  __syncthreads();

  // batch-mix: features[e] rows are L2-resident (84MB < 192MB L2)
  for (int idx = tid; idx < 4096; idx += 256) {
    const int c = idx >> 6, j = idx & 63;
    const float* fp = features + (size_t)c * HW + xy0 + j;
    float s0 = 0.f, s1 = 0.f, s2 = 0.f;
    for (int e = 0; e < 20; ++e) {
      const float f = fp[(size_t)e * (CC * HW)];
      s0 += attn_s[0][e] * f; s1 += attn_s[1][e] * f; s2 += attn_s[2][e] * f;
    }
    fmix[0][c][j] = s0; fmix[1][c][j] = s1; fmix[2][c][j] = s2;
  }
  __syncthreads();

  const int lane = tid & 31, wv = tid >> 5, lm = lane & 15, half = lane >> 4;
  float ps1 = 0.f, ps2 = 0.f;
#pragma unroll
  for (int tt = 0; tt < 2; ++tt) {
    const int t = wv + tt * 8;         // 16 tiles, 8 waves x 2
    const int ot = t >> 2, jt = t & 3;
    v8f_ acc = {};
    for (int h = 0; h < 3; ++h)
      acc = wmma64(WA + (h * 64 + ot * 16) * 64, 64,
                   &fmix[h][0][jt * 16], 64, acc, lm, half);
#pragma unroll
    for (int r = 0; r < 8; ++r) {
      const int o = ot * 16 + r + 8 * half;
      const int j = jt * 16 + lm;
      float v = acc[r] + b1[o];
      v = gelu_exact(v);
      const size_t gi = (size_t)(b * 64 + o) * HW + xy0 + j;
      v += features[gi];
      X[gi] = v;
      ps1 += v; ps2 += v * v;
    }
  }
  rs[tid] = ps1; rq[tid] = ps2; __syncthreads();
  for (int off = 128; off; off >>= 1) {
    if (tid < off) { rs[tid] += rs[tid + off]; rq[tid] += rq[tid + off]; }
    __syncthreads();
  }
  if (tid == 0) {
    partials[blockIdx.x * 2]     = rs[0];
    partials[blockIdx.x * 2 + 1] = rq[0];
  }
}

// ---------------------------------------------------------------------------
// Kernel 6: deterministic reduce of 5120 block partials -> (mean, rstd)
// ---------------------------------------------------------------------------
__global__ __launch_bounds__(256) void reduce_stats_kernel(const float* __restrict__ partials,
                                                           int n, float invN,
                                                           float* __restrict__ statsOut) {
  __shared__ float rs[256], rq[256];
  const int tid = threadIdx.x;
  float s1 = 0.f, s2 = 0.f;
  for (int i = tid; i < n; i += 256) { s1 += partials[2 * i]; s2 += partials[2 * i + 1]; }
  rs[tid] = s1; rq[tid] = s2; __syncthreads();
  for (int off = 128; off; off >>= 1) {
    if (tid < off) { rs[tid] += rs[tid + off]; rq[tid] += rq[tid + off]; }
    __syncthreads();
  }
  if (tid == 0) {
    const float m = rs[0] * invN;
    const float var = rq[0] * invN - m * m;
    statsOut[0] = m;
    statsOut[1] = rsqrtf(var + EPSLN);
  }
}

// ---------------------------------------------------------------------------
// Kernel 7: out_ = LN1(x1)*g1+beta1 (tile->LDS), out2 = W2@out_+b2 (WMMA),
//           x2 = gelu(out2)+out_  -> X in place ; partial sums for LN2.
// ---------------------------------------------------------------------------
__global__ __launch_bounds__(256) void fused2_kernel(const float* __restrict__ g1,
                                                     const float* __restrict__ beta1,
                                                     const float* __restrict__ W2,
                                                     const float* __restrict__ b2,
                                                     const float* __restrict__ stats,
                                                     float* __restrict__ X,
                                                     float* __restrict__ partials) {
  __shared__ float t_in[64][64];       // 16 KB : out_ tile
  __shared__ float rs[256], rq[256];
  const int tid = threadIdx.x;
  const int b   = blockIdx.x / 256;
  const int xy0 = (blockIdx.x % 256) * 64;
  const float m1 = stats[0], r1 = stats[1];

  for (int idx = tid; idx < 4096; idx += 256) {
    const int c = idx >> 6, j = idx & 63;
    const size_t gi = (size_t)(b * 64 + c) * HW + xy0 + j;
    const float xn = (X[gi] - m1) * r1;
    t_in[c][j] = xn * g1[gi] + beta1[gi];
  }
  __syncthreads();

  const int lane = tid & 31, wv = tid >> 5, lm = lane & 15, half = lane >> 4;
  float ps1 = 0.f, ps2 = 0.f;
#pragma unroll
  for (int tt = 0; tt < 2; ++tt) {
    const int t = wv + tt * 8;
    const int ot = t >> 2, jt = t & 3;
    v8f_ acc = {};
    acc = wmma64(W2 + (ot * 16) * 64, 64, &t_in[0][jt * 16], 64, acc, lm, half);
#pragma unroll
    for (int r = 0; r < 8; ++r) {
      const int o = ot * 16 + r + 8 * half;
      const int j = jt * 16 + lm;
      float v = acc[r] + b2[o];
      v = gelu_exact(v) + t_in[o][j];  // residual with out_
      const size_t gi = (size_t)(b * 64 + o) * HW + xy0 + j;
      X[gi] = v;                       // safe in-place: tile fully read above
      ps1 += v; ps2 += v * v;
    }
  }
  rs[tid] = ps1; rq[tid] = ps2; __syncthreads();
  for (int off = 128; off; off >>= 1) {
    if (tid < off) { rs[tid] += rs[tid + off]; rq[tid] += rq[tid + off]; }
    __syncthreads();
  }
  if (tid == 0) {
    partials[blockIdx.x * 2]     = rs[0];
    partials[blockIdx.x * 2 + 1] = rq[0];
  }
}

// ---------------------------------------------------------------------------
// Kernel 8: final LN2 elementwise -> d_out[NPOOL..]
// ---------------------------------------------------------------------------
__global__ __launch_bounds__(256) void ln2_out_kernel(const float* __restrict__ X,
                                                      const float* __restrict__ g2,
                                                      const float* __restrict__ beta2,
                                                      const float* __restrict__ stats2,
                                                      float* __restrict__ out) {
  const size_t i = (size_t)blockIdx.x * 256 + threadIdx.x;
  if (i < (size_t)NTOT) {
    const float m = stats2[0], r = stats2[1];
    out[i] = (X[i] - m) * r * g2[i] + beta2[i];
  }
}

// ---------------------------------------------------------------------------
// Kernel 9: 4x4 avg pool of `out` -> d_out[0..NPOOL)
// ---------------------------------------------------------------------------
__global__ __launch_bounds__(256) void pool_kernel(const float* __restrict__ out,
                                                   float* __restrict__ pooled_out) {
  const int t = blockIdx.x * 256 + threadIdx.x;
  if (t < NPOOL) {
    const int px = t & 31, py = (t >> 5) & 31, c = (t >> 10) & 63, b = t >> 16;
    const float* base = out + (size_t)(b * 64 + c) * HW + (py * 4) * WW + px * 4;
    float s = 0.f;
#pragma unroll
    for (int i = 0; i < 4; ++i)
#pragma unroll
      for (int j = 0; j < 4; ++j) s += base[i * WW + j];
    pooled_out[t] = s * (1.0f / 16.0f);
  }
}

// ---------------------------------------------------------------------------
// Host launch
// ---------------------------------------------------------------------------
extern "C" void kernel_launch(void* const* d_in, const int* in_sizes, int n_in,
                              void* d_out, int out_size, void* d_ws, size_t ws_size,
                              hipStream_t stream) {
  (void)in_sizes; (void)n_in; (void)out_size; (void)ws_size;
  const float* pooled   = (const float*)d_in[0];
  const float* features = (const float*)d_in[1];
  const float* Wq       = (const float*)d_in[2];
  const float* Wk       = (const float*)d_in[3];
  const float* Wv       = (const float*)d_in[4];
  const float* W1       = (const float*)d_in[5];
  const float* b1       = (const float*)d_in[6];
  const float* W2       = (const float*)d_in[7];
  const float* b2       = (const float*)d_in[8];
  const float* g1       = (const float*)d_in[9];
  const float* beta1    = (const float*)d_in[10];
  const float* g2       = (const float*)d_in[11];
  const float* beta2    = (const float*)d_in[12];

  float* out_full   = (float*)d_out + NPOOL;  // tuple order: (pooled_out, out)
  float* pooled_out = (float*)d_out;

  // workspace layout (floats) ~115.4 MB total
  float* ws       = (float*)d_ws;
  float* qbuf     = ws;                       // 3,932,160
  float* kbuf     = qbuf + 3932160;           // 3,932,160
  float* scores   = kbuf + 3932160;           // 1,200
  float* attn     = scores + 1200;            // 1,200
  float* WA       = attn + 1200;              // 12,288
  float* stats    = WA + 12288;               // 4 (m1,rstd1,m2,rstd2)
  float* partials = stats + 8;                // 10,240
  float* X        = partials + 10240;         // 20,971,520 (x1 then x2 in place)

  const float invN = 1.0f / (float)NTOT;

  qk_kernel<<<dim3(3840), dim3(256), 0, stream>>>(pooled, Wq, Wk, qbuf, kbuf);
  scores_kernel<<<dim3(1200), dim3(256), 0, stream>>>(qbuf, kbuf, scores);
  softmax_kernel<<<dim3(60), dim3(32), 0, stream>>>(scores, attn);
  wa_kernel<<<dim3(48), dim3(256), 0, stream>>>(W1, Wv, WA);
  fused1_kernel<<<dim3(BB * 256), dim3(256), 0, stream>>>(features, attn, WA, b1, X, partials);
  reduce_stats_kernel<<<dim3(1), dim3(256), 0, stream>>>(partials, BB * 256, invN, stats);
  fused2_kernel<<<dim3(BB * 256), dim3(256), 0, stream>>>(g1, beta1, W2, b2, stats, X, partials);
  reduce_stats_kernel<<<dim3(1), dim3(256), 0, stream>>>(partials, BB * 256, invN, stats + 2);
  ln2_out_kernel<<<dim3((NTOT + 255) / 256), dim3(256), 0, stream>>>(X, g2, beta2, stats + 2, out_full);
  pool_kernel<<<dim3((NPOOL + 255) / 256), dim3(256), 0, stream>>>(out_full, pooled_out);
}